// Entropy_Hist_26637387169825
// MI455X (gfx1250) — compile-verified
//
#include <hip/hip_runtime.h>
#include <hip/hip_bf16.h>

// Problem constants (from the reference)
#define HH 256
#define WW 256
#define BATCH 8
#define CH 128
#define ROWS (BATCH * CH)            // 1024 per-(b,c) images
#define TILE_H 16
#define TILES_PER_IMG (HH / TILE_H)  // 16
#define TILE_ROWS (TILE_H + 2)       // 18 rows incl. halo
#define TPB 256
#define BINS 256
#define K_SEL 64                     // RATIO * C = 0.5 * 128
#define HW (HH * WW)                 // 65536

typedef __attribute__((ext_vector_type(2))) float v2f;
typedef __attribute__((ext_vector_type(8))) float v8f;
typedef int v4i __attribute__((vector_size(16)));   // matches builtin proto

// CDNA5 async global->LDS path (ASYNCcnt), guarded so a toolchain without the
// builtin falls back to load+ds_store.
#if defined(__has_builtin)
#if __has_builtin(__builtin_amdgcn_global_load_async_to_lds_b128) && \
    __has_builtin(__builtin_amdgcn_s_wait_asynccnt)
#define USE_ASYNC_LDS 1
#endif
#endif

#ifdef USE_ASYNC_LDS
#define AS_GLOBAL __attribute__((address_space(1)))
#define AS_SHARED __attribute__((address_space(3)))
#endif

// ---------------------------------------------------------------------------
// Issue the loads for one 18x256 tile (16 rows + 1-row halo top/bottom,
// zero-padded outside the image) into an LDS buffer. 18*64 = 1152 float4 units.
// Async version: one global_load_async_to_lds_b128 per unit (no VGPR staging).
// ---------------------------------------------------------------------------
__device__ __forceinline__ void tile_issue(const float* __restrict__ base,
                                           float (*buf)[WW], int y0, int tid) {
  for (int i = tid; i < TILE_ROWS * 64; i += TPB) {
    const int lr = i >> 6;        // LDS row 0..17
    const int c4 = i & 63;        // float4 column
    const int gy = y0 - 1 + lr;   // image row
    float* ldsp = &buf[lr][c4 * 4];
    if (gy >= 0 && gy < HH) {
#ifdef USE_ASYNC_LDS
      __builtin_amdgcn_global_load_async_to_lds_b128(
          (AS_GLOBAL v4i*)(base + (size_t)gy * WW + c4 * 4),
          (AS_SHARED v4i*)ldsp, 0, 0);
#else
      *(float4*)ldsp = ((const float4*)(base + (size_t)gy * WW))[c4];
#endif
    } else {
      *(float4*)ldsp = make_float4(0.f, 0.f, 0.f, 0.f);  // zero padding (ds)
    }
  }
}

__device__ __forceinline__ void tile_wait() {
#ifdef USE_ASYNC_LDS
  __builtin_amdgcn_s_wait_asynccnt(0);
#endif
}

__device__ __forceinline__ float hsum3(const float (*tile)[WW], int r, int x) {
  const float c = tile[r][x];
  const float l = (x > 0) ? tile[r][x - 1] : 0.f;
  const float rv = (x < WW - 1) ? tile[r][x + 1] : 0.f;
  return l + c + rv;
}

// ---------------------------------------------------------------------------
// Pass 1: per-image min/max of ij = 100*center + (boxsum - center)/8
// Double-buffered: prefetch tile t+1 (async) while computing tile t.
// ---------------------------------------------------------------------------
__global__ __launch_bounds__(TPB) void k_minmax(const float* __restrict__ img,
                                                float* __restrict__ partials) {
  __shared__ float buf[2][TILE_ROWS][WW];   // 2 * 18KB
  __shared__ float red[TPB];
  const int row = blockIdx.x;               // (b,c) image id
  const int x = threadIdx.x;                // column 0..255
  const float* base = img + (size_t)row * HW;

  float mn = __FLT_MAX__;
  float mx = -__FLT_MAX__;

  tile_issue(base, buf[0], 0, x);
  tile_wait();
  __syncthreads();

  for (int t = 0; t < TILES_PER_IMG; ++t) {
    if (t + 1 < TILES_PER_IMG) {
      tile_issue(base, buf[(t + 1) & 1], (t + 1) * TILE_H, x);
    }
    const float(*tile)[WW] = buf[t & 1];

    float hp = hsum3(tile, 0, x);
    float hc = hsum3(tile, 1, x);
#pragma unroll 4
    for (int rr = 1; rr <= TILE_H; ++rr) {
      const float hn = hsum3(tile, rr + 1, x);
      const float ctr = tile[rr][x];
      const float ij = ctr * 100.0f + (hp + hc + hn - ctr) * 0.125f;
      mn = fminf(mn, ij);
      mx = fmaxf(mx, ij);
      hp = hc;
      hc = hn;
    }
    tile_wait();       // next tile resident in LDS
    __syncthreads();   // all waves done reading current buffer
  }

  // block reduction: min
  red[x] = mn;
  __syncthreads();
  for (int s = TPB / 2; s > 0; s >>= 1) {
    if (x < s) red[x] = fminf(red[x], red[x + s]);
    __syncthreads();
  }
  if (x == 0) partials[2 * row + 0] = red[0];
  __syncthreads();
  // block reduction: max
  red[x] = mx;
  __syncthreads();
  for (int s = TPB / 2; s > 0; s >>= 1) {
    if (x < s) red[x] = fmaxf(red[x], red[x + s]);
    __syncthreads();
  }
  if (x == 0) partials[2 * row + 1] = red[0];
}

// ---------------------------------------------------------------------------
// Reduce 1024 per-image (min,max) pairs to the global (mn, mx)
// ---------------------------------------------------------------------------
__global__ __launch_bounds__(256) void k_reduce_minmax(
    const float* __restrict__ partials, float* __restrict__ mnmx) {
  __shared__ float smn[256];
  __shared__ float smx[256];
  const int t = threadIdx.x;
  float mn = __FLT_MAX__, mx = -__FLT_MAX__;
  for (int i = t; i < ROWS; i += 256) {
    mn = fminf(mn, partials[2 * i + 0]);
    mx = fmaxf(mx, partials[2 * i + 1]);
  }
  smn[t] = mn;
  smx[t] = mx;
  __syncthreads();
  for (int s = 128; s > 0; s >>= 1) {
    if (t < s) {
      smn[t] = fminf(smn[t], smn[t + s]);
      smx[t] = fmaxf(smx[t], smx[t + s]);
    }
    __syncthreads();
  }
  if (t == 0) {
    mnmx[0] = smn[0];
    mnmx[1] = smx[0];
  }
}

// ---------------------------------------------------------------------------
// Pass 2: per-image 256-bin histogram (LDS atomics, exact integer counts ->
// deterministic) followed by entropy. The 256-value entropy sum uses
// V_WMMA_F32_16X16X4_F32 on wave 0 (A = h-values 16x16 fed as 4 chunks of K=4,
// B = all-ones -> row sums), then a lane-16 shuffle combines the halves.
// ---------------------------------------------------------------------------
__global__ __launch_bounds__(TPB) void k_hist_entropy(
    const float* __restrict__ img, const float* __restrict__ mnmx,
    float* __restrict__ ent) {
  __shared__ float buf[2][TILE_ROWS][WW];
  __shared__ unsigned int hist[BINS];
  __shared__ float hv[BINS];

  const int row = blockIdx.x;
  const int x = threadIdx.x;
  const float* base = img + (size_t)row * HW;

  hist[x] = 0u;
  const float mn = mnmx[0];
  const float mx = mnmx[1];
  const float rcp = 1.0f / (mx - mn);

  tile_issue(base, buf[0], 0, x);
  tile_wait();
  __syncthreads();   // tile 0 + hist init visible

  for (int t = 0; t < TILES_PER_IMG; ++t) {
    if (t + 1 < TILES_PER_IMG) {
      tile_issue(base, buf[(t + 1) & 1], (t + 1) * TILE_H, x);
    }
    const float(*tile)[WW] = buf[t & 1];

    float hp = hsum3(tile, 0, x);
    float hc = hsum3(tile, 1, x);
#pragma unroll 4
    for (int rr = 1; rr <= TILE_H; ++rr) {
      const float hn = hsum3(tile, rr + 1, x);
      const float ctr = tile[rr][x];
      const float ij = ctr * 100.0f + (hp + hc + hn - ctr) * 0.125f;
      // bin = clamp(floor((ij - mn)/(mx - mn) * 256), 0, 255)
      const float tn = (ij - mn) * rcp * 256.0f;
      int b = (int)floorf(tn);
      b = min(max(b, 0), BINS - 1);
      atomicAdd(&hist[b], 1u);
      hp = hc;
      hc = hn;
    }
    tile_wait();
    __syncthreads();
  }

  // h[t] = -p * log(max(p,1e-40)) / log(2),  p = count / ((W+2)*(H+2))
  {
    const float cnt = (float)hist[x];
    const float p = cnt / 66564.0f;   // 258*258
    const float pc = fmaxf(p, 1e-40f);
    hv[x] = (-p * logf(pc)) / 0.69314718055994531f;
  }
  __syncthreads();

  // Wave 0: ent[row] = sum_{t<256} hv[t] via V_WMMA_F32_16X16X4_F32.
  // A (16x4 f32) layout: lanes 0-15 hold (K=0,K=1), lanes 16-31 hold (K=2,K=3)
  // for row M = lane&15. B = all-ones 4x16 => D[m,n] = sum_k A[m,k].
  if (x < 32) {
    v2f bmat;
    bmat.x = 1.0f;
    bmat.y = 1.0f;
    v8f acc = {0.f, 0.f, 0.f, 0.f, 0.f, 0.f, 0.f, 0.f};
    const int m = x & 15;
    const int ko = (x < 16) ? 0 : 2;
#pragma unroll
    for (int chv = 0; chv < 4; ++chv) {
      v2f amat;
      amat.x = hv[m * 16 + chv * 4 + ko + 0];
      amat.y = hv[m * 16 + chv * 4 + ko + 1];
      acc = __builtin_amdgcn_wmma_f32_16x16x4_f32(
          false, amat, false, bmat, (short)0, acc, false, false);
    }
    // D layout: VGPR r, lanes 0-15 -> M=r, lanes 16-31 -> M=r+8 (any column).
    float part = acc[0] + acc[1] + acc[2] + acc[3] +
                 acc[4] + acc[5] + acc[6] + acc[7];
    part += __shfl_xor(part, 16, 32);   // combine M=0..7 and M=8..15 halves
    if (x == 0) ent[row] = part;
  }
}

// ---------------------------------------------------------------------------
// Stable descending top-k per batch via rank counting (matches lax.top_k:
// ties broken toward the lower index).
// ---------------------------------------------------------------------------
__global__ __launch_bounds__(CH) void k_topk(const float* __restrict__ ent,
                                             int* __restrict__ sel) {
  __shared__ float e[CH];
  const int b = blockIdx.x;
  const int i = threadIdx.x;   // channel
  e[i] = ent[b * CH + i];
  __syncthreads();
  const float v = e[i];
  int rank = 0;
  for (int j = 0; j < CH; ++j) {
    const float u = e[j];
    rank += (u > v) || (u == v && j < i);
  }
  if (rank < K_SEL) sel[b * K_SEL + rank] = i;
}

// ---------------------------------------------------------------------------
// Gather selected channels: out[b, j] = img[b, sel[b*64+j]]  (float4 copies)
// Grid: 8*64 images * 8 chunks = 4096 blocks, 256 threads, 8 float4 each.
// ---------------------------------------------------------------------------
__global__ __launch_bounds__(256) void k_gather(const float* __restrict__ img,
                                                const int* __restrict__ sel,
                                                float* __restrict__ out) {
  const int bi = blockIdx.x;
  const int chunk = bi & 7;
  const int im = bi >> 3;           // 0..511 output image id
  const int b = im >> 6;
  const int j = im & 63;
  const int c = sel[b * K_SEL + j];

  const float4* __restrict__ src =
      (const float4*)(img + ((size_t)(b * CH + c)) * HW) + chunk * 2048;
  float4* __restrict__ dst = (float4*)(out + (size_t)im * HW) + chunk * 2048;

  const int t = threadIdx.x;
#pragma unroll
  for (int k = 0; k < 8; ++k) {
    if (k + 2 < 8) {
      __builtin_prefetch(&src[t + (k + 2) * 256], 0, 3);  // global_prefetch_b8
    }
    dst[t + k * 256] = src[t + k * 256];
  }
}

// ---------------------------------------------------------------------------
// Launch
// ---------------------------------------------------------------------------
extern "C" void kernel_launch(void* const* d_in, const int* in_sizes, int n_in,
                              void* d_out, int out_size, void* d_ws,
                              size_t ws_size, hipStream_t stream) {
  const float* img = (const float*)d_in[0];   // [8,128,256,256] f32
  float* out = (float*)d_out;                 // [8,64,256,256] f32

  float* f = (float*)d_ws;
  float* mnmx = f;                          // 2 floats
  float* partials = f + 16;                 // 2*1024 floats
  float* ent = f + 16 + 2 * ROWS;           // 1024 floats
  int* sel = (int*)(f + 16 + 2 * ROWS + ROWS);  // 512 ints (~14.5 KB total)

  k_minmax<<<ROWS, TPB, 0, stream>>>(img, partials);
  k_reduce_minmax<<<1, 256, 0, stream>>>(partials, mnmx);
  k_hist_entropy<<<ROWS, TPB, 0, stream>>>(img, mnmx, ent);
  k_topk<<<BATCH, CH, 0, stream>>>(ent, sel);
  k_gather<<<BATCH * K_SEL * 8, 256, 0, stream>>>(img, sel, out);
}